// GraphNN_KNN_v1_v0_27384711479656
// MI455X (gfx1250) — compile-verified
//
#include <hip/hip_runtime.h>
#include <hip/hip_bf16.h>

typedef __attribute__((ext_vector_type(2))) float v2f;
typedef __attribute__((ext_vector_type(8))) float v8f;

#define N_NODES   500000
#define N_EDGES   8000000
#define N_ORDERS  4
#define EPO       2000000          // edges per order
#define TILES_PER_ORDER (EPO / 16) // 125000
#define TILES_PER_WAVE  8
#define D_NODE    10

// ---------------------------------------------------------------------------
// Prep: M1 = W_mp[:,0:10] - W_mp[:,10:20], M2 = W_mp[:,10:20], cv = W_mp[:,20]
// ---------------------------------------------------------------------------
__global__ void prep_weights(const float* __restrict__ Wmp,
                             float* __restrict__ M1,
                             float* __restrict__ M2,
                             float* __restrict__ cv) {
    int t = threadIdx.x;
    if (t < 100) {
        int r = t / 10, c = t % 10;
        float wa = Wmp[r * 21 + c];
        float wb = Wmp[r * 21 + 10 + c];
        M1[r * 10 + c] = wa - wb;
        M2[r * 10 + c] = wb;
    } else if (t < 110) {
        int r = t - 100;
        cv[r] = Wmp[r * 21 + 20];
    }
}

// ---------------------------------------------------------------------------
// Edge message-passing kernel (transposed roles):
//   D(16x16) = [M1 | c | 0](16ch x 12) @ [Xi ; dist ; 0](12 x 16edges)
//            + [M2 | 0 | 0](16ch x 12) @ [Xj ; 0 ; 0](12 x 16edges)  + b_mp
// A (weights) is loop-invariant; lane owns edge n = lane%16 in B and in D,
// so the destination node index for the scatter is the lane's own register.
// ---------------------------------------------------------------------------
__global__ __launch_bounds__(256) void edge_mp_wmma(
    const int*   __restrict__ order,   // this order's 2M indices into edges
    const int*   __restrict__ ei,      // edge_index, (2, N_EDGES) row-major
    const float* __restrict__ dist,    // (N_EDGES,)
    const float* __restrict__ M1,
    const float* __restrict__ M2,
    const float* __restrict__ cv,
    const float* __restrict__ bmp,
    const float* __restrict__ Xc,      // read (pre-update x)
    float*       __restrict__ Xn)      // accumulate (x + aggr)
{
    const int lane = threadIdx.x & 31;
    const int wave = (blockIdx.x * blockDim.x + threadIdx.x) >> 5;
    const int h    = lane >> 4;   // half-wave: K pair select / channel group
    const int el   = lane & 15;   // A row (channel) AND B col (edge slot)

    // ---- A fragments: weight rows, loop-invariant.
    // el >= 10 reads stay inside d_ws (finite poison); rows m>=10 never stored.
    const float* m1 = M1 + el * 10;
    const float* m2 = M2 + el * 10;
    v2f a1[3], a2[3];
    a1[0] = *(const v2f*)(m1 + 2 * h);
    a1[1] = *(const v2f*)(m1 + 4 + 2 * h);
    a2[0] = *(const v2f*)(m2 + 2 * h);
    a2[1] = *(const v2f*)(m2 + 4 + 2 * h);
    {
        v2f t1 = *(const v2f*)(m1 + 8);
        v2f t2 = *(const v2f*)(m2 + 8);
        float cvv = cv[el];                       // in-ws for el<16
        a1[2].x = h ? cvv : t1.x;                 // K=10 carries dist coeff
        a1[2].y = h ? 0.f : t1.y;
        a2[2].x = h ? 0.f : t2.x;
        a2[2].y = h ? 0.f : t2.y;
    }

    // ---- C init: D row m = v + 8h gets bias b_mp[m] ----
    float ci[8];
    #pragma unroll
    for (int v = 0; v < 8; ++v) {
        int m = v + 8 * h;
        ci[v] = (m < D_NODE) ? bmp[m] : 0.f;
    }

    const int tile0 = wave * TILES_PER_WAVE;
    #pragma unroll 1
    for (int t = 0; t < TILES_PER_WAVE; ++t) {
        const int tile = tile0 + t;
        if (tile >= TILES_PER_ORDER) break;       // wave-uniform guard

        const int   e  = order[tile * 16 + el];
        const int   i  = ei[e];
        const int   j  = ei[N_EDGES + e];
        const float dd = dist[e];
        const float* xi = Xc + (size_t)i * D_NODE;
        const float* xj = Xc + (size_t)j * D_NODE;

        // ---- B fragments: features of own edge, branch-free ----
        v2f b1[3], b2[3];
        b1[0] = *(const v2f*)(xi + 2 * h);
        b1[1] = *(const v2f*)(xi + 4 + 2 * h);
        b2[0] = *(const v2f*)(xj + 2 * h);
        b2[1] = *(const v2f*)(xj + 4 + 2 * h);
        v2f t1 = *(const v2f*)(xi + 8);           // k={8,9}: always in-bounds
        v2f t2 = *(const v2f*)(xj + 8);
        b1[2].x = h ? dd  : t1.x;                 // K=10 carries dist
        b1[2].y = h ? 0.f : t1.y;
        b2[2].x = h ? 0.f : t2.x;
        b2[2].y = h ? 0.f : t2.y;

        v8f C = {ci[0], ci[1], ci[2], ci[3], ci[4], ci[5], ci[6], ci[7]};
        #pragma unroll
        for (int s = 0; s < 3; ++s)
            C = __builtin_amdgcn_wmma_f32_16x16x4_f32(false, a1[s], false, b1[s],
                                                      (short)0, C, false, false);
        #pragma unroll
        for (int s = 0; s < 3; ++s)
            C = __builtin_amdgcn_wmma_f32_16x16x4_f32(false, a2[s], false, b2[s],
                                                      (short)0, C, false, false);

        // ---- scatter: lane's own edge, channels m = v + 8h ----
        float* base = Xn + (size_t)i * D_NODE + 8 * h;
        unsafeAtomicAdd(base + 0, C[0]);          // ch 8h+0
        unsafeAtomicAdd(base + 1, C[1]);          // ch 8h+1
        if (h == 0) {                             // single divergent branch
            #pragma unroll
            for (int v = 2; v < 8; ++v)
                unsafeAtomicAdd(base + v, C[v]);  // ch 2..7
        }
    }
}

// ---------------------------------------------------------------------------
// Final: out(500000x10) = X @ W_out^T + b_out  (same transposed mapping,
// channels contiguous per lane -> b64 stores)
// ---------------------------------------------------------------------------
__global__ __launch_bounds__(256) void out_gemm_wmma(
    const float* __restrict__ X,
    const float* __restrict__ Wout,   // (10,10) row-major
    const float* __restrict__ bout,
    float*       __restrict__ out)
{
    const int lane = threadIdx.x & 31;
    const int wave = (blockIdx.x * blockDim.x + threadIdx.x) >> 5;
    const int h    = lane >> 4;
    const int el   = lane & 15;       // A row (channel) AND B col (node row)

    const int row0 = wave * 16;
    if (row0 >= N_NODES) return;      // wave-uniform (500000 % 16 == 0)

    // A = W_out rows; guarded (input buffer is exactly 100 floats)
    v2f a[3] = {{0.f, 0.f}, {0.f, 0.f}, {0.f, 0.f}};
    if (el < D_NODE) {
        const float* w = Wout + el * 10;
        a[0] = *(const v2f*)(w + 2 * h);
        a[1] = *(const v2f*)(w + 4 + 2 * h);
        if (h == 0) a[2] = *(const v2f*)(w + 8);
    }
    float ci[8];
    #pragma unroll
    for (int v = 0; v < 8; ++v) {
        int m = v + 8 * h;
        ci[v] = (m < D_NODE) ? bout[m] : 0.f;
    }

    // B = node-feature columns (lane's own row)
    const int row = row0 + el;
    const float* xr = X + (size_t)row * D_NODE;
    v2f b[3];
    b[0] = *(const v2f*)(xr + 2 * h);
    b[1] = *(const v2f*)(xr + 4 + 2 * h);
    v2f t = *(const v2f*)(xr + 8);
    b[2].x = h ? 0.f : t.x;
    b[2].y = h ? 0.f : t.y;

    v8f C = {ci[0], ci[1], ci[2], ci[3], ci[4], ci[5], ci[6], ci[7]};
    #pragma unroll
    for (int s = 0; s < 3; ++s)
        C = __builtin_amdgcn_wmma_f32_16x16x4_f32(false, a[s], false, b[s],
                                                  (short)0, C, false, false);

    // channels v+8h of own row; pairs -> aligned b64 stores
    float* ob = out + (size_t)row * D_NODE + 8 * h;
    v2f p0; p0.x = C[0]; p0.y = C[1];
    *(v2f*)(ob) = p0;                 // ch 8h+0, 8h+1
    if (h == 0) {
        v2f p1; p1.x = C[2]; p1.y = C[3];
        v2f p2; p2.x = C[4]; p2.y = C[5];
        v2f p3; p3.x = C[6]; p3.y = C[7];
        *(v2f*)(ob + 2) = p1;
        *(v2f*)(ob + 4) = p2;
        *(v2f*)(ob + 6) = p3;
    }
}

// ---------------------------------------------------------------------------
extern "C" void kernel_launch(void* const* d_in, const int* in_sizes, int n_in,
                              void* d_out, int out_size, void* d_ws, size_t ws_size,
                              hipStream_t stream) {
    const float* x      = (const float*)d_in[0];
    const int*   ei     = (const int*)  d_in[1];
    const int*   orders = (const int*)  d_in[2];
    const float* dst    = (const float*)d_in[3];
    const float* W_mp   = (const float*)d_in[4];
    const float* b_mp   = (const float*)d_in[5];
    const float* W_out  = (const float*)d_in[6];
    const float* b_out  = (const float*)d_in[7];
    float* out = (float*)d_out;

    float* wsf = (float*)d_ws;
    float* M1  = wsf;          // 100 floats
    float* M2  = wsf + 128;    // 100 floats
    float* cv  = wsf + 256;    // 10 floats (+ poison up to 512 is read-safe)
    float* X0  = wsf + 512;                         // 5,000,000 floats
    float* X1  = X0 + (size_t)N_NODES * D_NODE;     // 5,000,000 floats

    prep_weights<<<1, 128, 0, stream>>>(W_mp, M1, M2, cv);

    const size_t xbytes = (size_t)N_NODES * D_NODE * sizeof(float);
    hipMemcpyAsync(X0, x, xbytes, hipMemcpyDeviceToDevice, stream);

    // 15632 waves cover 125000 tiles (8 tiles/wave), 8 waves/block
    const int edge_blocks = 1954;
    float* cur = X0;
    float* nxt = X1;
    for (int k = 0; k < N_ORDERS; ++k) {
        hipMemcpyAsync(nxt, cur, xbytes, hipMemcpyDeviceToDevice, stream);
        edge_mp_wmma<<<edge_blocks, 256, 0, stream>>>(
            orders + (size_t)k * EPO, ei, dst, M1, M2, cv, b_mp, cur, nxt);
        float* tmp = cur; cur = nxt; nxt = tmp;
    }

    // 31250 tiles of 16 rows, 8 waves/block -> 3907 blocks
    out_gemm_wmma<<<3907, 256, 0, stream>>>(cur, W_out, b_out, out);
}